// JP_Featurization_3332894621749
// MI455X (gfx1250) — compile-verified
//
#include <hip/hip_runtime.h>
#include <math.h>

#define OUT_FEATS 64
#define N_HEADS   4
#define CLIP_EPS  0.001f
#define PI_F      3.14159265358979323846f

typedef __attribute__((ext_vector_type(2))) float v2f;
typedef __attribute__((ext_vector_type(8))) float v8f;

// Fast reciprocal: v_rcp_f32 (1 ulp). Inputs here are exact small integers
// (segment counts), so this is effectively exact for the mean scaling.
__device__ __forceinline__ float fast_rcp(float x) {
    return __builtin_amdgcn_rcpf(x);
}

// ---------------------------------------------------------------------------
// Kernel 1: per line-graph edge. Compute symmetry key + spatial weights and
// scatter into per-g-edge sufficient statistics S[g][key*4+h] (16 floats/edge).
// ---------------------------------------------------------------------------
__global__ void lg_edge_kernel(const int* __restrict__ an,
                               const int* __restrict__ g_src,
                               const int* __restrict__ g_dst,
                               const int* __restrict__ lg_src,
                               const int* __restrict__ lg_dst,
                               const float* __restrict__ costheta,
                               const float* __restrict__ dnr,
                               const float* __restrict__ pa,
                               const float* __restrict__ pb,
                               const float* __restrict__ pc,
                               const float* __restrict__ pd,
                               float* __restrict__ S,
                               float* __restrict__ cnt_lg,
                               int Elg)
{
    int e = blockIdx.x * blockDim.x + threadIdx.x;
    if (e >= Elg) return;

    int g = lg_src[e];
    int j = lg_dst[e];
    int ka = an[g_src[g]];   // k_src[lg_src]
    int kb = an[g_dst[g]];   // k_dst[lg_src]
    int kc = an[g_dst[j]];   // k_dst[lg_dst]
    int peripheral = (ka == kc) ? 1 : 0;
    int central    = ((kb == ka) && (kb == kc)) ? 1 : 0;
    int key = peripheral * 2 + central;

    float ct    = fminf(fmaxf(costheta[e], -CLIP_EPS), CLIP_EPS);
    float theta = acosf(ct);
    float dn    = dnr[e];
    float dn2   = dn * dn;

    float* dst = S + (size_t)g * 16 + key * 4;
#pragma unroll
    for (int h = 0; h < 4; ++h) {
        float Bm  = fmodf(pb[h], PI_F);
        float ang = powf((cosf(pa[h] * theta + Bm) + 1.0f) * 0.5f, pc[h]);
        float rad = expf(-pd[h] * dn2);
        atomicAdd(dst + h, ang * rad);
    }
    atomicAdd(cnt_lg + g, 1.0f);
}

// ---------------------------------------------------------------------------
// Kernel 2: per g-edge. Apply lg-segment mean scale and scatter the 16-float
// statistic into per-node accumulators T[n][k*4+h]; count g-edges per node.
// ---------------------------------------------------------------------------
__global__ void edge_to_node_kernel(const int* __restrict__ g_src,
                                    const float* __restrict__ S,
                                    const float* __restrict__ cnt_lg,
                                    float* __restrict__ T,
                                    float* __restrict__ cnt_g,
                                    int Eg)
{
    int i = blockIdx.x * blockDim.x + threadIdx.x;
    if (i >= Eg) return;

    int   n     = g_src[i];
    float scale = fast_rcp(fmaxf(cnt_lg[i], 1.0f));

    const float4* sp = (const float4*)(S + (size_t)i * 16);
    float* tp = T + (size_t)n * 16;
#pragma unroll
    for (int q = 0; q < 4; ++q) {
        float4 v = sp[q];
        atomicAdd(tp + q * 4 + 0, v.x * scale);
        atomicAdd(tp + q * 4 + 1, v.y * scale);
        atomicAdd(tp + q * 4 + 2, v.z * scale);
        atomicAdd(tp + q * 4 + 3, v.w * scale);
    }
    atomicAdd(cnt_g + n, 1.0f);
}

// ---------------------------------------------------------------------------
// Kernel 3: dense stage. out[n,f] = (1/cnt_g[n]) * sum_{K=0..15} T[n,K]*W[K,f]
// with W[k*4+h, f] = VT[k, f*4+h]. One wave handles a 16-node tile; K=16 is
// done as 4 chained V_WMMA_F32_16X16X4_F32 ops (one per symmetry key) for each
// of the 4 feature tiles -> 16 WMMAs per wave, full f32 precision.
//
// A layout (f32 16x4): lanes 0-15 M=lane, K={0,1} in (v0,v1); lanes 16-31
// M=lane-16, K={2,3}. B layout (f32 4x16): lanes 0-15 N=lane rows {0,1};
// lanes 16-31 N=lane-16 rows {2,3}. C/D: vgpr r holds M=r (lanes<16) /
// M=r+8 (lanes>=16), N=lane&15.
// ---------------------------------------------------------------------------
__global__ void gemm_out_kernel(const float* __restrict__ T,
                                const float* __restrict__ cnt_g,
                                const float* __restrict__ VT,
                                float* __restrict__ out,
                                int N, int ntiles)
{
    int lane = threadIdx.x & 31;
    int wid  = threadIdx.x >> 5;
    int tile = blockIdx.x * (blockDim.x >> 5) + wid;
    if (tile >= ntiles) return;            // wave-uniform: EXEC stays all-ones

    int  node_base = tile * 16;
    bool full_tile = (node_base + 16 <= N);  // wave-uniform
    int  m         = lane & 15;
    int  krow      = (lane < 16) ? 0 : 2;  // which K rows this half-wave holds

    // A operand, reused across all 4 feature tiles.
    int node   = node_base + m;
    int node_c = node < N ? node : (N - 1);
    v2f Av[4];
#pragma unroll
    for (int kk = 0; kk < 4; ++kk) {
        const float* tp = T + (size_t)node_c * 16 + kk * 4 + krow;
        Av[kk].x = tp[0];
        Av[kk].y = tp[1];
    }

    // Per-row reciprocal counts for the final segment-mean (v_rcp_f32; counts
    // are exact integers so approximation error is negligible).
    float inv[8];
#pragma unroll
    for (int r = 0; r < 8; ++r) {
        int row = node_base + ((lane < 16) ? r : r + 8);
        int rc  = row < N ? row : (N - 1);
        inv[r] = fast_rcp(fmaxf(cnt_g[rc], 1.0f));
    }

    int col = lane & 15;
#pragma unroll
    for (int ft = 0; ft < 4; ++ft) {
        int f = ft * 16 + col;
        v8f acc = {};
#pragma unroll
        for (int kk = 0; kk < 4; ++kk) {
            // B[row, col] for K-chunk kk: VT[kk, f*4 + row], row = krow/krow+1
            const float* bp = VT + kk * (OUT_FEATS * N_HEADS) + f * 4 + krow;
            v2f Bv;
            Bv.x = bp[0];
            Bv.y = bp[1];
            acc = __builtin_amdgcn_wmma_f32_16x16x4_f32(
                false, Av[kk], false, Bv, (short)0, acc, false, false);
        }
        if (full_tile) {
            // Common path: no per-element predication, no exec churn.
#pragma unroll
            for (int r = 0; r < 8; ++r) {
                int row = node_base + ((lane < 16) ? r : r + 8);
                out[(size_t)row * OUT_FEATS + f] = acc[r] * inv[r];
            }
        } else {
#pragma unroll
            for (int r = 0; r < 8; ++r) {
                int row = node_base + ((lane < 16) ? r : r + 8);
                if (row < N)
                    out[(size_t)row * OUT_FEATS + f] = acc[r] * inv[r];
            }
        }
    }
}

// ---------------------------------------------------------------------------
extern "C" void kernel_launch(void* const* d_in, const int* in_sizes, int n_in,
                              void* d_out, int out_size, void* d_ws, size_t ws_size,
                              hipStream_t stream)
{
    const int*   an       = (const int*)d_in[0];
    const int*   g_src    = (const int*)d_in[1];
    const int*   g_dst    = (const int*)d_in[2];
    const int*   lg_src   = (const int*)d_in[3];
    const int*   lg_dst   = (const int*)d_in[4];
    const float* costheta = (const float*)d_in[5];
    const float* dnr      = (const float*)d_in[6];
    const float* pa       = (const float*)d_in[7];
    const float* pb       = (const float*)d_in[8];
    const float* pc       = (const float*)d_in[9];
    const float* pd       = (const float*)d_in[10];
    const float* VT       = (const float*)d_in[11];

    int N   = in_sizes[0];
    int Eg  = in_sizes[1];
    int Elg = in_sizes[3];

    // Workspace layout (floats): S[Eg*16] | cnt_lg[Eg] | T[N*16] | cnt_g[N]
    float* S      = (float*)d_ws;
    float* cnt_lg = S + (size_t)Eg * 16;
    float* T      = cnt_lg + Eg;
    float* cnt_g  = T + (size_t)N * 16;
    size_t ws_needed = ((size_t)Eg * 17 + (size_t)N * 17) * sizeof(float);
    hipMemsetAsync(d_ws, 0, ws_needed, stream);

    lg_edge_kernel<<<(Elg + 255) / 256, 256, 0, stream>>>(
        an, g_src, g_dst, lg_src, lg_dst, costheta, dnr,
        pa, pb, pc, pd, S, cnt_lg, Elg);

    edge_to_node_kernel<<<(Eg + 255) / 256, 256, 0, stream>>>(
        g_src, S, cnt_lg, T, cnt_g, Eg);

    int ntiles = (N + 15) / 16;
    gemm_out_kernel<<<(ntiles + 3) / 4, 128, 0, stream>>>(
        T, cnt_g, VT, (float*)d_out, N, ntiles);
}